// GATLayer_9912784519915
// MI455X (gfx1250) — compile-verified
//
#include <hip/hip_runtime.h>

#define IN_DIM 256
#define OUT_DIM 64

typedef float v2f __attribute__((ext_vector_type(2)));
typedef float v8f __attribute__((ext_vector_type(8)));

// ---------------------------------------------------------------------------
// Kernel 1: z = h @ W_fc^T  via V_WMMA_F32_16X16X4_F32
// One wave computes a 16x64 tile of z (4 accumulators of 16x16).
// A (16x4 f32): lanes 0-15 -> M=lane, VGPR0=K0,VGPR1=K1; lanes 16-31 -> K2,K3.
// B (4x16 f32): lanes 0-15 -> N=lane, VGPR0=K0,VGPR1=K1; lanes 16-31 -> K2,K3.
// C (16x16 f32): VGPR r, lanes 0-15 -> M=r, lanes 16-31 -> M=8+r.
// ---------------------------------------------------------------------------
__global__ __launch_bounds__(128) void k_gemm_z(const float* __restrict__ h,
                                                const float* __restrict__ Wfc,
                                                float* __restrict__ z,
                                                int n_nodes) {
  const int wave = (blockIdx.x * blockDim.x + threadIdx.x) >> 5;
  const int lane = threadIdx.x & 31;
  const int m0 = wave * 16;
  if (m0 >= n_nodes) return;              // wave-uniform: EXEC stays all-1s

  const int half = lane >> 4;             // 0: K=0,1   1: K=2,3
  const int l16  = lane & 15;

  v8f c0 = {}, c1 = {}, c2 = {}, c3 = {};

  const float* __restrict__ hrow = h + (size_t)(m0 + l16) * IN_DIM;
  const float* __restrict__ w0 = Wfc + (size_t)(0 * 16 + l16) * IN_DIM;
  const float* __restrict__ w1 = Wfc + (size_t)(1 * 16 + l16) * IN_DIM;
  const float* __restrict__ w2 = Wfc + (size_t)(2 * 16 + l16) * IN_DIM;
  const float* __restrict__ w3 = Wfc + (size_t)(3 * 16 + l16) * IN_DIM;

  for (int k = 0; k < IN_DIM; k += 4) {
    const int ka = k + half * 2;
    v2f a;  a.x = hrow[ka]; a.y = hrow[ka + 1];
    v2f b0; b0.x = w0[ka];  b0.y = w0[ka + 1];
    v2f b1; b1.x = w1[ka];  b1.y = w1[ka + 1];
    v2f b2; b2.x = w2[ka];  b2.y = w2[ka + 1];
    v2f b3; b3.x = w3[ka];  b3.y = w3[ka + 1];
    c0 = __builtin_amdgcn_wmma_f32_16x16x4_f32(false, a, false, b0, (short)0, c0, false, false);
    c1 = __builtin_amdgcn_wmma_f32_16x16x4_f32(false, a, false, b1, (short)0, c1, false, false);
    c2 = __builtin_amdgcn_wmma_f32_16x16x4_f32(false, a, false, b2, (short)0, c2, false, false);
    c3 = __builtin_amdgcn_wmma_f32_16x16x4_f32(false, a, false, b3, (short)0, c3, false, false);
  }

  // Store 16x64 tile: VGPR r -> row m0 + half*8 + r, col j*16 + l16
  float* __restrict__ zb = z + (size_t)(m0 + half * 8) * OUT_DIM + l16;
#pragma unroll
  for (int r = 0; r < 8; ++r) {
    zb[(size_t)r * OUT_DIM + 0 * 16] = c0[r];
    zb[(size_t)r * OUT_DIM + 1 * 16] = c1[r];
    zb[(size_t)r * OUT_DIM + 2 * 16] = c2[r];
    zb[(size_t)r * OUT_DIM + 3 * 16] = c3[r];
  }
}

// ---------------------------------------------------------------------------
// Kernel 2: per-node attn scalars a_src/a_dst + init of m/denom/out.
// One wave per node; lanes cover the 64-dim row (2 elems each), xor-reduce.
// ---------------------------------------------------------------------------
__global__ __launch_bounds__(128) void k_node_attn(const float* __restrict__ z,
                                                   const float* __restrict__ Wattn,
                                                   float* __restrict__ a_src,
                                                   float* __restrict__ a_dst,
                                                   unsigned* __restrict__ m_u,
                                                   float* __restrict__ denom,
                                                   float* __restrict__ out,
                                                   int n_nodes) {
  const int node = (blockIdx.x * blockDim.x + threadIdx.x) >> 5;
  const int lane = threadIdx.x & 31;
  if (node >= n_nodes) return;

  const float* __restrict__ zr = z + (size_t)node * OUT_DIM;
  const float z0 = zr[lane], z1 = zr[lane + 32];
  float ps = z0 * Wattn[lane]      + z1 * Wattn[lane + 32];
  float pd = z0 * Wattn[64 + lane] + z1 * Wattn[96 + lane];
#pragma unroll
  for (int off = 16; off > 0; off >>= 1) {
    ps += __shfl_xor(ps, off, 32);
    pd += __shfl_xor(pd, off, 32);
  }
  if (lane == 0) {
    a_src[node] = ps;
    a_dst[node] = pd;
    m_u[node]   = 0u;      // below fkey(x) for every non-NaN x
    denom[node] = 0.0f;
  }
  out[(size_t)node * OUT_DIM + lane]      = 0.0f;
  out[(size_t)node * OUT_DIM + lane + 32] = 0.0f;
}

// Order-preserving float<->uint key for atomicMax on floats of either sign.
__device__ __forceinline__ unsigned fkey(float f) {
  unsigned u = __float_as_uint(f);
  return (u & 0x80000000u) ? ~u : (u | 0x80000000u);
}
__device__ __forceinline__ float funkey(unsigned k) {
  return __uint_as_float((k & 0x80000000u) ? (k ^ 0x80000000u) : ~k);
}

// ---------------------------------------------------------------------------
// Kernel 3: per-edge logit + segment max (thread per edge).
// ---------------------------------------------------------------------------
__global__ __launch_bounds__(256) void k_edge_logit(const int* __restrict__ src,
                                                    const int* __restrict__ dst,
                                                    const float* __restrict__ a_src,
                                                    const float* __restrict__ a_dst,
                                                    float* __restrict__ e_buf,
                                                    unsigned* __restrict__ m_u,
                                                    int n_edges) {
  const int i = blockIdx.x * blockDim.x + threadIdx.x;
  if (i >= n_edges) return;
  const int d = dst[i];
  float e = a_src[src[i]] + a_dst[d];
  e = (e > 0.0f) ? e : 0.01f * e;         // leaky_relu, slope 0.01
  e_buf[i] = e;
  atomicMax(&m_u[d], fkey(e));
}

// ---------------------------------------------------------------------------
// Kernel 4: stable exp + segment sum (thread per edge).
// ---------------------------------------------------------------------------
__global__ __launch_bounds__(256) void k_edge_exp(const int* __restrict__ dst,
                                                  const float* __restrict__ e_buf,
                                                  const unsigned* __restrict__ m_u,
                                                  float* __restrict__ ee_buf,
                                                  float* __restrict__ denom,
                                                  int n_edges) {
  const int i = blockIdx.x * blockDim.x + threadIdx.x;
  if (i >= n_edges) return;
  const int d = dst[i];
  const float ee = __expf(e_buf[i] - funkey(m_u[d]));
  ee_buf[i] = ee;
  atomicAdd(&denom[d], ee);
}

// ---------------------------------------------------------------------------
// Kernel 5: out[dst] += alpha * z[src]  (wave per edge, 2 dims per lane).
// z and out are L2-resident (12.8 MB each in 192 MB L2).
// ---------------------------------------------------------------------------
__global__ __launch_bounds__(256) void k_edge_scatter(const int* __restrict__ src,
                                                      const int* __restrict__ dst,
                                                      const float* __restrict__ z,
                                                      const float* __restrict__ ee_buf,
                                                      const float* __restrict__ denom,
                                                      float* __restrict__ out,
                                                      int n_edges) {
  const int wave = (blockIdx.x * blockDim.x + threadIdx.x) >> 5;
  const int lane = threadIdx.x & 31;
  if (wave >= n_edges) return;
  const int s = src[wave], d = dst[wave];
  const float alpha = ee_buf[wave] / fmaxf(denom[d], 1e-16f);
  const float* __restrict__ zs = z + (size_t)s * OUT_DIM;
  float* __restrict__ ob = out + (size_t)d * OUT_DIM;
  atomicAdd(&ob[lane],      alpha * zs[lane]);
  atomicAdd(&ob[lane + 32], alpha * zs[lane + 32]);
}

// ---------------------------------------------------------------------------
extern "C" void kernel_launch(void* const* d_in, const int* in_sizes, int n_in,
                              void* d_out, int out_size, void* d_ws, size_t ws_size,
                              hipStream_t stream) {
  const float* h     = (const float*)d_in[0];
  const float* Wfc   = (const float*)d_in[1];
  const float* Wattn = (const float*)d_in[2];
  const int*   src   = (const int*)d_in[3];
  const int*   dst   = (const int*)d_in[4];
  float*       out   = (float*)d_out;

  const int n_nodes = in_sizes[0] / IN_DIM;   // 50000
  const int n_edges = in_sizes[3];            // 800000

  // Workspace layout (all 4-byte types)
  char* ws = (char*)d_ws;
  float*    z      = (float*)ws;                                  ws += (size_t)n_nodes * OUT_DIM * 4;
  float*    a_src  = (float*)ws;                                  ws += (size_t)n_nodes * 4;
  float*    a_dst  = (float*)ws;                                  ws += (size_t)n_nodes * 4;
  unsigned* m_u    = (unsigned*)ws;                               ws += (size_t)n_nodes * 4;
  float*    denom  = (float*)ws;                                  ws += (size_t)n_nodes * 4;
  float*    e_buf  = (float*)ws;                                  ws += (size_t)n_edges * 4;
  float*    ee_buf = (float*)ws;

  // 1) z = h @ Wfc^T  (WMMA f32)
  {
    const int tiles  = (n_nodes + 15) / 16;       // 3125
    const int blocks = (tiles + 3) / 4;           // 4 waves / block
    k_gemm_z<<<blocks, 128, 0, stream>>>(h, Wfc, z, n_nodes);
  }
  // 2) per-node scalars + init
  {
    const int blocks = (n_nodes + 3) / 4;         // 4 waves / block
    k_node_attn<<<blocks, 128, 0, stream>>>(z, Wattn, a_src, a_dst, m_u, denom, out, n_nodes);
  }
  // 3) logits + segment max
  k_edge_logit<<<(n_edges + 255) / 256, 256, 0, stream>>>(src, dst, a_src, a_dst, e_buf, m_u, n_edges);
  // 4) exp + segment sum
  k_edge_exp<<<(n_edges + 255) / 256, 256, 0, stream>>>(dst, e_buf, m_u, ee_buf, denom, n_edges);
  // 5) weighted scatter
  k_edge_scatter<<<(n_edges + 7) / 8, 256, 0, stream>>>(src, dst, z, ee_buf, denom, out, n_edges);
}